// DynamicConv1D_30648886624828
// MI455X (gfx1250) — compile-verified
//
#include <hip/hip_runtime.h>

typedef __attribute__((ext_vector_type(2))) float v2f;
typedef __attribute__((ext_vector_type(4))) float v4f;
typedef __attribute__((ext_vector_type(8))) float v8f;
typedef __attribute__((ext_vector_type(4))) int   v4i;

// address-space-qualified helper types for the async-copy builtin
typedef __attribute__((address_space(1))) void  as1_void;
typedef __attribute__((address_space(3))) void  as3_void;
typedef __attribute__((address_space(1))) v4i   as1_v4i;
typedef __attribute__((address_space(3))) v4i   as3_v4i;

#define B_    4
#define S_    2048
#define C_    1024
#define H_    16
#define D_    64
#define K_    7
#define PAD_  3
#define NOUT  112          // H_ * K_
#define KBS   116          // padded LDS row stride (floats) for kernel buffer
#define XROWS 22           // 16 positions + 2*PAD halo
#define XST   1028         // padded LDS row stride (floats) for x tile (bank spread)

// ---- CDNA5 async global->LDS copy (ASYNCcnt path), 16B per lane -------------------
__device__ __forceinline__ void async_copy16(const float* g, float* l) {
#if defined(__has_builtin) && __has_builtin(__builtin_amdgcn_global_load_async_to_lds_b128)
    __builtin_amdgcn_global_load_async_to_lds_b128(
        (as1_v4i*)(as1_void*)g,
        (as3_v4i*)(as3_void*)l, 0, 0);
#else
    asm volatile("global_load_async_to_lds_b128 %0, %1, off"
                 :: "v"((unsigned)(size_t)l), "v"((unsigned long long)(size_t)g)
                 : "memory");
#endif
}

__device__ __forceinline__ void wait_asynccnt0() {
#if defined(__has_builtin) && __has_builtin(__builtin_amdgcn_s_wait_asynccnt)
    __builtin_amdgcn_s_wait_asynccnt(0);
#else
    asm volatile("s_wait_asynccnt 0x0" ::: "memory");
#endif
}

// One block = 16 sequence positions of one batch element.
//   Phase 0 (all waves): issue async DMA of the 22-row x halo tile into LDS
//   Phase 1 (waves 0..6): 16x112 kernel-prediction GEMM via V_WMMA_F32_16X16X4_F32
//                         (A read from global -> overlaps with the async DMA)
//   Phase 2 (all 8 waves): 7-tap dynamic conv reading x from LDS, branch-free
__global__ __launch_bounds__(256) void dynconv_fused_kernel(
    const float* __restrict__ x,      // [B,S,C]
    const float* __restrict__ Wp,     // [C, 112]
    const float* __restrict__ bp,     // [112]
    float* __restrict__ out)          // [B,S,C]
{
    __shared__ float xs[XROWS * XST];  // staged x halo tile: rows s0-3 .. s0+18
    __shared__ float kbuf[16 * KBS];   // predicted kernels for this tile: [16 pos][112]

    const int blk  = blockIdx.x;
    const int b    = blk >> 7;            // / (S_/16)
    const int s0   = (blk & 127) << 4;    // tile start position

    const int tid  = threadIdx.x;
    const int wave = tid >> 5;
    const int lane = tid & 31;
    const int half = lane >> 4;           // 0: lanes 0-15, 1: lanes 16-31
    const int ln   = lane & 15;

    // ---------------- Phase 0: async-stage x halo tile into LDS -----------------------
    // 22 rows x 4KB; one async b128 op per lane covers 512B per wave -> 8 segs/row.
    // chunk q in [0,176): row r = q/8, segment seg = q%8; wave w takes q = w, w+8, ...
    for (int q = wave; q < XROWS * 8; q += 8) {
        const int r   = q >> 3;
        const int seg = q & 7;
        const int sp  = s0 + r - PAD_;
        float* ldst = &xs[r * XST + seg * 128 + lane * 4];
        if (sp >= 0 && sp < S_) {
            const float* gsrc = x + ((size_t)b * S_ + (size_t)sp) * C_ + seg * 128 + lane * 4;
            async_copy16(gsrc, ldst);
        } else {
            *(v4f*)ldst = (v4f){0.f, 0.f, 0.f, 0.f};  // zero-pad outside [0,S)
        }
    }

    // ---------------- Phase 1: GEMM  kernels = x_tile(16x1024) * Wp(1024x112) ---------
    if (wave < 7) {
        const int n0 = wave * 16;         // this wave's 16 output columns

        // A fragment: lane holds row M=ln; VGPR0/1 = K = 2*half, 2*half+1 (ISA 7.12.2)
        const float* xrow = x + ((size_t)b * S_ + (size_t)(s0 + ln)) * C_ + (2 * half);
        // B fragment: lane holds col N=ln; VGPR0/1 = K rows 2*half, 2*half+1
        const float* wcol = Wp + (size_t)(2 * half) * NOUT + (n0 + ln);

        v8f acc = {0.f, 0.f, 0.f, 0.f, 0.f, 0.f, 0.f, 0.f};

#pragma unroll 8
        for (int k0 = 0; k0 < C_; k0 += 4) {
            v2f a = *(const v2f*)(xrow + k0);              // x[s0+ln][k0+2h .. +1]
            v2f bb;
            bb.x = wcol[(size_t)(k0)     * NOUT];          // Wp[k0+2h  ][n0+ln]
            bb.y = wcol[(size_t)(k0 + 1) * NOUT];          // Wp[k0+2h+1][n0+ln]
            // 8-arg form: (neg_a, A, neg_b, B, c_mod, C, reuse_a, reuse_b)
            acc = __builtin_amdgcn_wmma_f32_16x16x4_f32(
                false, a, false, bb, (short)0, acc, false, false);
        }

        // C/D layout: VGPR r, lanes 0-15 -> M=r ; lanes 16-31 -> M=r+8 ; N=ln
        const float bias = bp[n0 + ln];
#pragma unroll
        for (int r = 0; r < 8; ++r) {
            const int m = r + 8 * half;
            kbuf[m * KBS + n0 + ln] = acc[r] + bias;
        }
    }

    // All async DMA issued by this wave must have landed before the block-wide barrier.
    wait_asynccnt0();
    __syncthreads();

    // ---------------- Phase 2: dynamic 7-tap conv (x from LDS, branch-free) -----------
    // thread tid owns channel group c = 4*tid (one float4, within a single head)
    const int c = tid << 2;               // 0..1020
    const int h = c >> 6;                 // head index (D_=64)

    float* ob = out + (size_t)b * S_ * C_;

#pragma unroll
    for (int p = 0; p < 16; ++p) {
        const float* kw = &kbuf[p * KBS + h * K_];
        v4f accv = {0.f, 0.f, 0.f, 0.f};
#pragma unroll
        for (int i = 0; i < K_; ++i) {
            // LDS row (p+i) corresponds to sequence position s0 + p + i - PAD_
            const v4f xv = *(const v4f*)(&xs[(p + i) * XST + c]);
            const float w = kw[i];
            accv.x = fmaf(w, xv.x, accv.x);
            accv.y = fmaf(w, xv.y, accv.y);
            accv.z = fmaf(w, xv.z, accv.z);
            accv.w = fmaf(w, xv.w, accv.w);
        }
        *(v4f*)(ob + (size_t)(s0 + p) * C_ + c) = accv;
    }
}

extern "C" void kernel_launch(void* const* d_in, const int* in_sizes, int n_in,
                              void* d_out, int out_size, void* d_ws, size_t ws_size,
                              hipStream_t stream) {
    const float* x  = (const float*)d_in[0];   // [4,2048,1024] fp32
    const float* Wp = (const float*)d_in[1];   // [1024,112]    fp32
    const float* bp = (const float*)d_in[2];   // [112]         fp32
    float* out      = (float*)d_out;           // [4,2048,1024] fp32

    (void)in_sizes; (void)n_in; (void)out_size; (void)d_ws; (void)ws_size;

    const int blocks = B_ * (S_ / 16);         // 512
    dynconv_fused_kernel<<<blocks, 256, 0, stream>>>(x, Wp, bp, out);
}